// ContextAwareGraphPooling_6030134083774
// MI455X (gfx1250) — compile-verified
//
#include <hip/hip_runtime.h>

#define NN 16384
#define CC 32

typedef __attribute__((ext_vector_type(2))) float v2f;
typedef __attribute__((ext_vector_type(8))) float v8f;

// ---------------------------------------------------------------------------
// K1: attention MLP -> per-node score.  Weights are wave-uniform -> s_loads.
// ---------------------------------------------------------------------------
__global__ void __launch_bounds__(256)
k_scores(const float* __restrict__ x,
         const float* __restrict__ w1, const float* __restrict__ b1,
         const float* __restrict__ w2, const float* __restrict__ b2,
         const float* __restrict__ w3, const float* __restrict__ b3,
         float* __restrict__ scores) {
    int n = blockIdx.x * blockDim.x + threadIdx.x;
    float xr[32];
#pragma unroll
    for (int k = 0; k < 32; ++k) xr[k] = x[n * 32 + k];
    float h2[64];
#pragma unroll
    for (int m = 0; m < 64; ++m) h2[m] = b2[m];
    for (int j = 0; j < 128; ++j) {
        float a = b1[j];
#pragma unroll
        for (int k = 0; k < 32; ++k) a = fmaf(xr[k], w1[k * 128 + j], a);
        a = fmaxf(a, 0.0f);
#pragma unroll
        for (int m = 0; m < 64; ++m) h2[m] = fmaf(a, w2[j * 64 + m], h2[m]);
    }
    float s = b3[0];
#pragma unroll
    for (int m = 0; m < 64; ++m) s = fmaf(fmaxf(h2[m], 0.0f), w3[m], s);
    scores[n] = s;
}

// ---------------------------------------------------------------------------
// K2: softmax statistics (max, sum of exp) over all N scores. One block.
// ---------------------------------------------------------------------------
__global__ void __launch_bounds__(1024)
k_softmax_stats(const float* __restrict__ scores, float* __restrict__ stats) {
    __shared__ float red[1024];
    int t = threadIdx.x;
    float m = -3.0e38f;
    for (int i = t; i < NN; i += 1024) m = fmaxf(m, scores[i]);
    red[t] = m; __syncthreads();
    for (int s = 512; s > 0; s >>= 1) { if (t < s) red[t] = fmaxf(red[t], red[t + s]); __syncthreads(); }
    float mx = red[0]; __syncthreads();
    float sum = 0.0f;
    for (int i = t; i < NN; i += 1024) sum += expf(scores[i] - mx);
    red[t] = sum; __syncthreads();
    for (int s = 512; s > 0; s >>= 1) { if (t < s) red[t] += red[t + s]; __syncthreads(); }
    if (t == 0) { stats[0] = mx; stats[1] = red[0]; }
}

// ---------------------------------------------------------------------------
// K3: attn[n] = softmax(scores)[n]; per-block partial gc = sum_n x[n,:]*attn[n].
// Deterministic fixed-order reduction (no float atomics).
// ---------------------------------------------------------------------------
__global__ void __launch_bounds__(256)
k_attn_gc(const float* __restrict__ x, const float* __restrict__ scores,
          const float* __restrict__ stats,
          float* __restrict__ attn, float* __restrict__ gcpart) {
    __shared__ float lds[256 * 32];
    int t = threadIdx.x;
    int n = blockIdx.x * 256 + t;
    float a = expf(scores[n] - stats[0]) * (1.0f / stats[1]);
    attn[n] = a;
#pragma unroll
    for (int c = 0; c < 32; ++c) lds[t * 32 + c] = x[n * 32 + c] * a;
    __syncthreads();
    if (t < 32) {
        float s = 0.0f;
        for (int r = 0; r < 256; ++r) s += lds[r * 32 + t];
        gcpart[blockIdx.x * 32 + t] = s;
    }
}

__global__ void k_gc_reduce(const float* __restrict__ gcpart, float* __restrict__ gc) {
    int c = threadIdx.x;  // 32 threads
    float s = 0.0f;
    for (int b = 0; b < 64; ++b) s += gcpart[b * 32 + c];
    gc[c] = s;
}

// ---------------------------------------------------------------------------
// K5: context partials via WMMA F32 16x16x4.
// Each wave owns a 16-row tile of AW = A @ wf (wf = x*attn recomputed on the
// fly), accumulating over K = all 16384 columns, then dots AW with wf.
// A-operand: lane l -> row i0+(l&15), K pair j+2*(l>>4)+{0,1}  => b64 stream.
// C/D layout: reg r -> row i0 + r + 8*(l>>4), col (l&15) [+16 for acc1].
// ---------------------------------------------------------------------------
__global__ void __launch_bounds__(256)
k_context(const float* __restrict__ A, const float* __restrict__ x,
          const float* __restrict__ attn, float* __restrict__ ctxpart) {
    __shared__ float red[256];
    int t = threadIdx.x;
    int wave = t >> 5, lane = t & 31;
    int half = lane >> 4, lm = lane & 15;
    int i0 = blockIdx.x * 128 + wave * 16;

    const float* Arow = A + (size_t)(i0 + lm) * NN + 2 * half;
    const float* attp = attn + 2 * half;

    v8f acc0 = {0.f, 0.f, 0.f, 0.f, 0.f, 0.f, 0.f, 0.f};
    v8f acc1 = {0.f, 0.f, 0.f, 0.f, 0.f, 0.f, 0.f, 0.f};

#pragma unroll 4
    for (int j = 0; j < NN; j += 4) {
        v2f av = *(const v2f*)(Arow + j);           // A[i0+lm, j+2h .. +1]
        v2f at = *(const v2f*)(attp + j);           // attn[j+2h], attn[j+2h+1]
        int rB = j + 2 * half;
        float x00 = x[rB * 32 + lm];
        float x01 = x[(rB + 1) * 32 + lm];
        float x10 = x[rB * 32 + lm + 16];
        float x11 = x[(rB + 1) * 32 + lm + 16];
        v2f b0 = {x00 * at.x, x01 * at.y};          // wf rows rB,rB+1 cols 0..15
        v2f b1 = {x10 * at.x, x11 * at.y};          // wf rows rB,rB+1 cols 16..31
        acc0 = __builtin_amdgcn_wmma_f32_16x16x4_f32(false, av, false, b0,
                                                     (short)0, acc0, false, false);
        acc1 = __builtin_amdgcn_wmma_f32_16x16x4_f32(false, av, false, b1,
                                                     (short)0, acc1, false, false);
    }

    // partial = sum over tile of AW .* wf
    float p = 0.0f;
#pragma unroll
    for (int r = 0; r < 8; ++r) {
        int row = i0 + r + 8 * half;
        float ar = attn[row];
        p = fmaf(acc0[r], x[row * 32 + lm] * ar, p);
        p = fmaf(acc1[r], x[row * 32 + lm + 16] * ar, p);
    }
    red[t] = p; __syncthreads();
    for (int s = 128; s > 0; s >>= 1) { if (t < s) red[t] += red[t + s]; __syncthreads(); }
    if (t == 0) ctxpart[blockIdx.x] = red[0];
}

__global__ void __launch_bounds__(128)
k_ctx_reduce(const float* __restrict__ ctxpart, const float* __restrict__ gc,
             float* __restrict__ gc2) {
    __shared__ float red[128];
    int t = threadIdx.x;
    red[t] = ctxpart[t]; __syncthreads();
    for (int s = 64; s > 0; s >>= 1) { if (t < s) red[t] += red[t + s]; __syncthreads(); }
    float ctx = red[0] * (1.0f / (float)NN);
    if (t < 32) gc2[t] = gc[t] + 0.001f * ctx;
}

// ---------------------------------------------------------------------------
// K7: global MLP 32 -> 1024 -> 1024 -> 256 (ReLU each).  block 0: gc, block 1: gc2.
// ---------------------------------------------------------------------------
__global__ void __launch_bounds__(1024)
k_mlp(const float* __restrict__ gc, const float* __restrict__ gc2,
      const float* __restrict__ w1, const float* __restrict__ b1,
      const float* __restrict__ w2, const float* __restrict__ b2,
      const float* __restrict__ w3, const float* __restrict__ b3,
      float* __restrict__ out2) {
    __shared__ float in_s[32];
    __shared__ float h1[1024];
    __shared__ float h2[1024];
    int t = threadIdx.x, b = blockIdx.x;
    const float* in = (b == 0) ? gc : gc2;
    if (t < 32) in_s[t] = in[t];
    __syncthreads();
    float a = b1[t];
#pragma unroll
    for (int k = 0; k < 32; ++k) a = fmaf(in_s[k], w1[k * 1024 + t], a);
    h1[t] = fmaxf(a, 0.0f);
    __syncthreads();
    a = b2[t];
#pragma unroll 8
    for (int k = 0; k < 1024; ++k) a = fmaf(h1[k], w2[k * 1024 + t], a);
    h2[t] = fmaxf(a, 0.0f);
    __syncthreads();
    if (t < 256) {
        a = b3[t];
#pragma unroll 8
        for (int k = 0; k < 1024; ++k) a = fmaf(h2[k], w3[k * 256 + t], a);
        out2[b * 256 + t] = fmaxf(a, 0.0f);
    }
}

__global__ void k_combine(const float* __restrict__ out2, float* __restrict__ out) {
    int t = threadIdx.x;
    float o = out2[t];
    float m = out2[256 + t];
    o = o + 0.001f * m;   // replicate the reference's two separate adds
    o = o + 0.001f * m;
    out[t] = o;
}

// ---------------------------------------------------------------------------
extern "C" void kernel_launch(void* const* d_in, const int* in_sizes, int n_in,
                              void* d_out, int out_size, void* d_ws, size_t ws_size,
                              hipStream_t stream) {
    const float* x    = (const float*)d_in[0];
    const float* adj  = (const float*)d_in[1];
    const float* g_w1 = (const float*)d_in[2];
    const float* g_b1 = (const float*)d_in[3];
    const float* g_w2 = (const float*)d_in[4];
    const float* g_b2 = (const float*)d_in[5];
    const float* g_w3 = (const float*)d_in[6];
    const float* g_b3 = (const float*)d_in[7];
    const float* a_w1 = (const float*)d_in[8];
    const float* a_b1 = (const float*)d_in[9];
    const float* a_w2 = (const float*)d_in[10];
    const float* a_b2 = (const float*)d_in[11];
    const float* a_w3 = (const float*)d_in[12];
    const float* a_b3 = (const float*)d_in[13];

    float* ws     = (float*)d_ws;
    float* scores = ws + 0;        // 16384
    float* attn   = ws + 16384;    // 16384
    float* stats  = ws + 32768;    // 2
    float* gcpart = ws + 32772;    // 64*32
    float* gc     = ws + 34820;    // 32
    float* ctxp   = ws + 34852;    // 128
    float* gc2    = ws + 34980;    // 32
    float* out2   = ws + 35012;    // 512

    k_scores<<<64, 256, 0, stream>>>(x, a_w1, a_b1, a_w2, a_b2, a_w3, a_b3, scores);
    k_softmax_stats<<<1, 1024, 0, stream>>>(scores, stats);
    k_attn_gc<<<64, 256, 0, stream>>>(x, scores, stats, attn, gcpart);
    k_gc_reduce<<<1, 32, 0, stream>>>(gcpart, gc);
    k_context<<<128, 256, 0, stream>>>(adj, x, attn, ctxp);
    k_ctx_reduce<<<1, 128, 0, stream>>>(ctxp, gc, gc2);
    k_mlp<<<2, 1024, 0, stream>>>(gc, gc2, g_w1, g_b1, g_w2, g_b2, g_w3, g_b3, out2);
    k_combine<<<1, 256, 0, stream>>>(out2, (float*)d_out);
}